// TODToSky_31138512896875
// MI455X (gfx1250) — compile-verified
//
#include <hip/hip_runtime.h>
#include <stdint.h>

#define NPIX   196608
#define CHUNK  256
#define NTHR   256
#define DEPTH  3          // pipeline stages in flight

// ---- CDNA5 async copy: 16B global -> LDS, tracked by ASYNCcnt ------------
__device__ __forceinline__ void async_copy16(uint32_t lds_off, uint64_t gaddr) {
    asm volatile("global_load_async_to_lds_b128 %0, %1, off"
                 :: "v"(lds_off), "v"(gaddr) : "memory");
}

__device__ __forceinline__ void atomic_add_f32(float* p, float v) {
    // lowers to global_atomic_add_f32 (no-return form; STOREcnt tracked)
    unsafeAtomicAdd(p, v);
}

// ---- scatter: sky[b, pix[s], k] += tod[b,s] * w[s,k] ---------------------
__global__ __launch_bounds__(NTHR) void scatter_kernel(
    const float* __restrict__ tod,      // [4, S]
    const float* __restrict__ weights,  // [S, 3]
    const int*   __restrict__ pix_idx,  // [S]
    float*       __restrict__ out,      // [4, NPIX, 3]
    int S)
{
    __shared__ __align__(16) int   sh_pix[DEPTH][CHUNK];        // 3 KB
    __shared__ __align__(16) float sh_w  [DEPTH][CHUNK * 3];    // 9 KB
    __shared__ __align__(16) float sh_tod[DEPTH][4 * CHUNK];    // 12 KB

    const int tid = threadIdx.x;

    const uint32_t pixBase = (uint32_t)(uintptr_t)&sh_pix[0][0];
    const uint32_t wBase   = (uint32_t)(uintptr_t)&sh_w  [0][0];
    const uint32_t todBase = (uint32_t)(uintptr_t)&sh_tod[0][0];

    const uint64_t gPix = (uint64_t)(uintptr_t)pix_idx;
    const uint64_t gW   = (uint64_t)(uintptr_t)weights;
    const uint64_t gTod = (uint64_t)(uintptr_t)tod;

    const long nfull = (long)S / CHUNK;     // full 256-sample chunks
    const long G     = (long)gridDim.x;
    const long c0    = (long)blockIdx.x;

    // One stage = 512 x 16B chunks: 64 (pix) + 192 (weights) + 256 (tod).
    // Thread t issues chunks 2t, 2t+1 -> each wave's 64-chunk span lies in
    // exactly one region: exactly 2 async b128 per wave per stage (uniform
    // ASYNCcnt accounting; EXEC==0 paths are execz-branched around).
    auto issue = [&](long c, int buf) {
        const long s0 = c * CHUNK;
        #pragma unroll
        for (int q = 0; q < 2; ++q) {
            const int ch = 2 * tid + q;
            uint64_t g; uint32_t l;
            if (ch < 64) {                       // pix_idx tile: 1 KB
                g = gPix + (uint64_t)(s0 * 4) + (uint64_t)(ch * 16);
                l = pixBase + (uint32_t)buf * (CHUNK * 4) + (uint32_t)(ch * 16);
            } else if (ch < 256) {               // weights tile: 3 KB
                const int cc = ch - 64;
                g = gW + (uint64_t)(s0 * 12) + (uint64_t)(cc * 16);
                l = wBase + (uint32_t)buf * (CHUNK * 12) + (uint32_t)(cc * 16);
            } else {                             // tod tile: 4 rows x 1 KB
                const int cc  = ch - 256;
                const int row = cc >> 6;
                const int col = cc & 63;
                g = gTod + ((uint64_t)row * (uint64_t)S + (uint64_t)s0) * 4
                         + (uint64_t)(col * 16);
                l = todBase + (uint32_t)buf * (4 * CHUNK * 4) + (uint32_t)(cc * 16);
            }
            async_copy16(l, g);
        }
    };

    // prologue: prime up to DEPTH stages
    if (c0 < nfull)         issue(c0,         0);
    if (c0 + G < nfull)     issue(c0 + G,     1);
    if (c0 + 2 * G < nfull) issue(c0 + 2 * G, 2);

    int buf = 0;
    for (long c = c0; c < nfull; c += G) {
        // wait until the CURRENT stage's 2 ops/wave have landed; stages
        // issued ahead of it (0..2) may remain outstanding (in-order done).
        if (c + G >= nfull) {
            asm volatile("s_wait_asynccnt 0x0" ::: "memory");
        } else if (c + 2 * G >= nfull) {
            asm volatile("s_wait_asynccnt 0x2" ::: "memory");
        } else {
            asm volatile("s_wait_asynccnt 0x4" ::: "memory");
        }
        __syncthreads();

        // consume: one sample per thread
        const int   pix = sh_pix[buf][tid];
        const float w0  = sh_w[buf][tid * 3 + 0];
        const float w1  = sh_w[buf][tid * 3 + 1];
        const float w2  = sh_w[buf][tid * 3 + 2];
        const unsigned pbase = (unsigned)pix * 3u;
        #pragma unroll
        for (int b = 0; b < 4; ++b) {
            const float t = sh_tod[buf][b * CHUNK + tid];
            float* o = out + (size_t)b * (size_t)(NPIX * 3) + pbase;
            atomic_add_f32(o + 0, t * w0);
            atomic_add_f32(o + 1, t * w1);
            atomic_add_f32(o + 2, t * w2);
        }

        __syncthreads();                 // buffer drained before refill
        if (c + (long)DEPTH * G < nfull) issue(c + (long)DEPTH * G, buf);
        if (++buf == DEPTH) buf = 0;
    }

    // tail (S % 256 == 128 samples) — direct loads by block 0
    if (c0 == 0) {
        const long s = nfull * CHUNK + tid;
        if (s < (long)S) {
            const int   pix = pix_idx[s];
            const float w0 = weights[s * 3 + 0];
            const float w1 = weights[s * 3 + 1];
            const float w2 = weights[s * 3 + 2];
            const unsigned pbase = (unsigned)pix * 3u;
            #pragma unroll
            for (int b = 0; b < 4; ++b) {
                const float t = tod[(size_t)b * (size_t)S + (size_t)s];
                float* o = out + (size_t)b * (size_t)(NPIX * 3) + pbase;
                atomic_add_f32(o + 0, t * w0);
                atomic_add_f32(o + 1, t * w1);
                atomic_add_f32(o + 2, t * w2);
            }
        }
    }
}

// ---- normalize: out[b, pix, :] /= cov[pix] (exact division, ref-match) ---
__global__ __launch_bounds__(NTHR) void normalize_kernel(
    float* __restrict__ out, const float* __restrict__ cov)
{
    const unsigned i = blockIdx.x * NTHR + threadIdx.x;   // over B*NPIX
    const unsigned total = 4u * (unsigned)NPIX;
    if (i >= total) return;
    const unsigned pix = i % (unsigned)NPIX;
    const float c = cov[pix];
    const size_t base = (size_t)i * 3;
    out[base + 0] = out[base + 0] / c;
    out[base + 1] = out[base + 1] / c;
    out[base + 2] = out[base + 2] / c;
}

extern "C" void kernel_launch(void* const* d_in, const int* in_sizes, int n_in,
                              void* d_out, int out_size, void* d_ws, size_t ws_size,
                              hipStream_t stream) {
    const float* tod     = (const float*)d_in[0];   // [B, S]
    const float* weights = (const float*)d_in[1];   // [S, 3]
    const float* cov     = (const float*)d_in[2];   // [NPIX]
    const int*   pix     = (const int*)  d_in[3];   // [S]
    float*       out     = (float*)d_out;           // [B, NPIX, 3]
    const int S = in_sizes[3];

    hipMemsetAsync(d_out, 0, (size_t)out_size * sizeof(float), stream);

    scatter_kernel<<<1024, NTHR, 0, stream>>>(tod, weights, pix, out, S);

    const unsigned total = 4u * (unsigned)NPIX;
    normalize_kernel<<<(total + NTHR - 1) / NTHR, NTHR, 0, stream>>>(out, cov);
}